// MahalanobisLoss_82875688943917
// MI455X (gfx1250) — compile-verified
//
#include <hip/hip_runtime.h>
#include <hip/hip_bf16.h>
#include <math.h>

typedef __attribute__((ext_vector_type(2))) float v2f;
typedef __attribute__((ext_vector_type(8))) float v8f;

#define D_DIM 256
#define PAD_ROWS 48              /* 3 spare 16-row tiles for tail strips */
#define EPS_NORM 1e-12f
#define EPS_COV  1e-6f
#define EPS_SQRT 1e-8f

#define WMMA_F32(Af, Bf, Cf) \
    __builtin_amdgcn_wmma_f32_16x16x4_f32(false, (Af), false, (Bf), (short)0, (Cf), false, false)

// ---------------------------------------------------------------------------
// Kernel 1: row L2-normalize. grid = (B, 2) — y-dim selects x vs y.
// ---------------------------------------------------------------------------
__global__ void mahal_normalize_kernel(const float* __restrict__ x,
                                       const float* __restrict__ y,
                                       float* __restrict__ xn,
                                       float* __restrict__ yn) {
    const float* src = (blockIdx.y == 0) ? x : y;
    float*       dst = (blockIdx.y == 0) ? xn : yn;
    const int row = blockIdx.x;
    const int t   = threadIdx.x;               // 0..255 == d

    float v = src[(size_t)row * D_DIM + t];

    __shared__ float red[D_DIM];
    red[t] = v * v;
    __syncthreads();
    #pragma unroll
    for (int s = 128; s > 0; s >>= 1) {
        if (t < s) red[t] += red[t + s];
        __syncthreads();
    }
    float nrm   = sqrtf(red[0]);
    float scale = 1.0f / fmaxf(nrm, EPS_NORM);
    dst[(size_t)row * D_DIM + t] = v * scale;
}

// ---------------------------------------------------------------------------
// Kernel 2: per-sample Az = (sigma_inv[label] + eps*I) @ zn  and  q = zn.Az
// One block (256 threads) per sample; zn row cached in LDS.
// ---------------------------------------------------------------------------
__global__ void mahal_matvec_kernel(const float* __restrict__ zn,
                                    const long long* __restrict__ labels,
                                    const float* __restrict__ sigma_inv,
                                    float* __restrict__ az,
                                    float* __restrict__ q) {
    const int row = blockIdx.x;                // sample index
    const int t   = threadIdx.x;               // output dim, 0..255

    __shared__ float zs[D_DIM];
    zs[t] = zn[(size_t)row * D_DIM + t];
    __syncthreads();

    const long long c = labels[row];
    const float* __restrict__ Arow =
        sigma_inv + ((size_t)c * D_DIM + t) * D_DIM;

    float acc = 0.f;
    #pragma unroll 8
    for (int k = 0; k < D_DIM; ++k)
        acc = fmaf(Arow[k], zs[k], acc);
    acc = fmaf(EPS_COV, zs[t], acc);           // + eps * I contribution

    az[(size_t)row * D_DIM + t] = acc;

    __shared__ float red[D_DIM];
    red[t] = acc * zs[t];
    __syncthreads();
    #pragma unroll
    for (int s = 128; s > 0; s >>= 1) {
        if (t < s) red[t] += red[t + s];
        __syncthreads();
    }
    if (t == 0) q[row] = red[0];
}

// ---------------------------------------------------------------------------
// Kernel 3: one wave per 16x64 strip (4 adjacent 16x16 J-tiles).
// J-subtiles are addressed off ONE base pointer per matrix via immediate
// offsets (u * 16 rows = 16 KB, fits the 24-bit IOFFSET) — arrays are padded
// by PAD_ROWS so tail strips read in-bounds garbage that the epilogue masks.
// The x-pass and y-pass run as separate k-loops (disjoint load sets, zero
// duplicated traffic) to keep VGPR pressure low enough to avoid spills.
// ---------------------------------------------------------------------------
__global__ __launch_bounds__(32, 1)
void mahal_pair_strip_kernel(const float* __restrict__ xn,
                             const float* __restrict__ axn,
                             const float* __restrict__ yn,
                             const float* __restrict__ ayn,
                             const float* __restrict__ qx,
                             const float* __restrict__ qy,
                             const long long* __restrict__ labels,
                             float* __restrict__ partials,
                             int T /* tiles per side = B/16 */) {
    // decode group id -> (ti, strip g): row ti has ceil((T-ti)/4) strips
    int rem = blockIdx.x;
    int ti  = 0;
    for (;;) {
        int g = (T - ti + 3) >> 2;
        if (rem < g) break;
        rem -= g;
        ++ti;
    }
    const int tjBase = ti + 4 * rem;

    const int lane = threadIdx.x;      // 0..31, wave32
    const int r    = lane & 15;        // fragment row (M%16) / column (N)
    const int kh   = lane >> 4;        // lane half: K sub-pair / M+8 select
    const int I    = ti * 16;
    const int J0   = tjBase * 16;

    v8f dx0 = {}, dx1 = {}, dx2 = {}, dx3 = {};
    v8f dy0 = {}, dy1 = {}, dy2 = {}, dy3 = {};

    // ---------------- x pass ----------------
    {
        const float* __restrict__ ZnI = xn  + (size_t)(I  + r) * D_DIM;
        const float* __restrict__ AzI = axn + (size_t)(I  + r) * D_DIM;
        const float* __restrict__ ZnJ = xn  + (size_t)(J0 + r) * D_DIM;
        const float* __restrict__ AzJ = axn + (size_t)(J0 + r) * D_DIM;

        #pragma unroll 2
        for (int k = 0; k < D_DIM; k += 4) {
            const int kb = k + kh * 2;
            v2f aZn = *(const v2f*)(ZnI + kb);
            v2f aAz = *(const v2f*)(AzI + kb);
#define XSTEP(u)                                                        \
            {                                                           \
                v2f bAz = *(const v2f*)(AzJ + kb + (u) * 16 * D_DIM);   \
                v2f bZn = *(const v2f*)(ZnJ + kb + (u) * 16 * D_DIM);   \
                dx##u = WMMA_F32(aZn, bAz, dx##u);                      \
                dx##u = WMMA_F32(aAz, bZn, dx##u);                      \
            }
            XSTEP(0) XSTEP(1) XSTEP(2) XSTEP(3)
#undef XSTEP
        }
    }

    // ---------------- y pass ----------------
    {
        const float* __restrict__ YnI = yn  + (size_t)(I  + r) * D_DIM;
        const float* __restrict__ AyI = ayn + (size_t)(I  + r) * D_DIM;
        const float* __restrict__ YnJ = yn  + (size_t)(J0 + r) * D_DIM;
        const float* __restrict__ AyJ = ayn + (size_t)(J0 + r) * D_DIM;

        #pragma unroll 2
        for (int k = 0; k < D_DIM; k += 4) {
            const int kb = k + kh * 2;
            v2f aYn = *(const v2f*)(YnI + kb);
            v2f aAy = *(const v2f*)(AyI + kb);
#define YSTEP(u)                                                        \
            {                                                           \
                v2f bAy = *(const v2f*)(AyJ + kb + (u) * 16 * D_DIM);   \
                v2f bYn = *(const v2f*)(YnJ + kb + (u) * 16 * D_DIM);   \
                dy##u = WMMA_F32(aYn, bAy, dy##u);                      \
                dy##u = WMMA_F32(aAy, bYn, dy##u);                      \
            }
            YSTEP(0) YSTEP(1) YSTEP(2) YSTEP(3)
#undef YSTEP
        }
    }

    // Epilogue: lane owns column j = tj*16 + (lane&15); rows i = I + rr + 8*kh
    float acc = 0.f;

#define EPI(u)                                                                \
    if (tjBase + (u) < T) {                                                   \
        const int j = (tjBase + (u)) * 16 + r;                                \
        const long long lj = labels[j];                                       \
        const float qxj = qx[j];                                              \
        const float qyj = qy[j];                                              \
        _Pragma("unroll")                                                     \
        for (int rr = 0; rr < 8; ++rr) {                                      \
            const int i = I + rr + 8 * kh;                                    \
            if (i < j && labels[i] == lj) {                                   \
                const float Qx = qx[i] + qxj - dx##u[rr];                     \
                const float Qy = qy[i] + qyj - dy##u[rr];                     \
                acc += fabsf(sqrtf(Qx + EPS_SQRT) - sqrtf(Qy + EPS_SQRT));    \
            }                                                                 \
        }                                                                     \
    }
    EPI(0) EPI(1) EPI(2) EPI(3)
#undef EPI

    // wave32 reduction
    #pragma unroll
    for (int off = 16; off > 0; off >>= 1)
        acc += __shfl_down(acc, off, 32);

    if (lane == 0) partials[blockIdx.x] = acc;
}

// ---------------------------------------------------------------------------
// Kernel 4: deterministic final reduction of strip partials, scale by 1/B.
// ---------------------------------------------------------------------------
__global__ void mahal_reduce_kernel(const float* __restrict__ partials,
                                    int n, float* __restrict__ out,
                                    float invB) {
    __shared__ float red[256];
    const int t = threadIdx.x;
    float acc = 0.f;
    for (int i = t; i < n; i += 256) acc += partials[i];
    red[t] = acc;
    __syncthreads();
    #pragma unroll
    for (int s = 128; s > 0; s >>= 1) {
        if (t < s) red[t] += red[t + s];
        __syncthreads();
    }
    if (t == 0) out[0] = red[0] * invB;
}

// ---------------------------------------------------------------------------
extern "C" void kernel_launch(void* const* d_in, const int* in_sizes, int n_in,
                              void* d_out, int out_size, void* d_ws, size_t ws_size,
                              hipStream_t stream) {
    const float*     x         = (const float*)d_in[0];
    const float*     y         = (const float*)d_in[1];
    const long long* labels    = (const long long*)d_in[2];   // int64 per reference
    const float*     sigma_inv = (const float*)d_in[3];
    float*           out       = (float*)d_out;

    const int d  = D_DIM;
    const int B  = in_sizes[0] / d;         // 1024
    const int BP = B + PAD_ROWS;            // padded row count per matrix
    const int T  = B / 16;                  // 64 tiles per side

    // number of 4-tile strips covering the upper triangle
    int ngroups = 0;
    for (int ti = 0; ti < T; ++ti) ngroups += (T - ti + 3) / 4;   // 544 for T=64

    // workspace layout (floats); xn/yn/axn/ayn padded by PAD_ROWS rows
    float* ws  = (float*)d_ws;
    float* xn  = ws;
    float* yn  = xn  + (size_t)BP * d;
    float* axn = yn  + (size_t)BP * d;
    float* ayn = axn + (size_t)BP * d;
    float* qx  = ayn + (size_t)BP * d;
    float* qy  = qx  + B;
    float* partials = qy + B;

    // 1) normalize rows of x and y
    mahal_normalize_kernel<<<dim3(B, 2), d, 0, stream>>>(x, y, xn, yn);

    // 2) per-sample A z and quadratic forms (x then y)
    mahal_matvec_kernel<<<B, d, 0, stream>>>(xn, labels, sigma_inv, axn, qx);
    mahal_matvec_kernel<<<B, d, 0, stream>>>(yn, labels, sigma_inv, ayn, qy);

    // 3) WMMA pair strips (one wave per 16x64 strip = 4 tiles)
    mahal_pair_strip_kernel<<<ngroups, 32, 0, stream>>>(
        xn, axn, yn, ayn, qx, qy, labels, partials, T);

    // 4) final deterministic reduction
    mahal_reduce_kernel<<<1, 256, 0, stream>>>(partials, ngroups, out,
                                               1.0f / (float)B);
}